// MotifConv2D_26388279066707
// MI455X (gfx1250) — compile-verified
//
#include <hip/hip_runtime.h>
#include <hip/hip_bf16.h>
#include <stdint.h>

// MotifConv2D (Go-pattern stencil), 4096x4096 int32 board -> 8 boolean maps (float32).
// Memory-bound: ~536 MB write-once output + 64 MB input => ~26us floor at 23.3 TB/s.
// CDNA5 strategy: async global->LDS b128 tile staging (ASYNCcnt path), LDS-fed
// 2x2 / cross stencil on a 4x2 per-thread micro-tile, b128 nontemporal stores.

#define HH 4096
#define WW 4096
#define OW 4095   // 2x2-motif output dimension
#define EW 4094   // eye output dimension

#define TX 128    // outputs per block in x
#define TY 16     // outputs per block in y
#define LROWS 18  // TY + 2 halo
#define LCOLS 132 // TX + 4 (33 int4 chunks per row; row = 528 B, 16B aligned)
#define NCHUNK (LROWS * 33) // 594 b128 transfers per block

#ifndef __has_builtin
#define __has_builtin(x) 0
#endif
#if __has_builtin(__builtin_amdgcn_global_load_async_to_lds_b128) && \
    __has_builtin(__builtin_amdgcn_s_wait_asynccnt)
#define USE_ASYNC_LDS 1
#endif

typedef int   v4i __attribute__((ext_vector_type(4)));
typedef float f4  __attribute__((ext_vector_type(4)));
typedef f4    uf4 __attribute__((aligned(4)));   // under-aligned quad for row-stride-4095 stores

__device__ __forceinline__ void stage_b128(const int* __restrict__ g, int* l) {
#if defined(USE_ASYNC_LDS)
    __builtin_amdgcn_global_load_async_to_lds_b128(
        (v4i*)(const_cast<int*>(g)), (v4i*)(l), /*imm offset*/ 0, /*cpol*/ 0);
#else
    *reinterpret_cast<int4*>(l) = *reinterpret_cast<const int4*>(g);
#endif
}

__global__ __launch_bounds__(256) void motif_stencil_kernel(
    const int* __restrict__ board, float* __restrict__ out) {
    __shared__ int tile[LROWS][LCOLS];

    const int tid = threadIdx.x;
    const int gx0 = blockIdx.x * TX;
    const int gy0 = blockIdx.y * TY;

    // ---- Stage (TY+2) x (TX+2) input halo into LDS with b128 async copies.
    // Index-clamped (no divergence): EXEC stays all-1s for the async path.
    // Clamped duplicates / out-of-board chunks only feed discarded outputs.
#pragma unroll
    for (int it = 0; it < 3; ++it) {
        int idx = tid + it * 256;
        idx = (idx < NCHUNK) ? idx : (NCHUNK - 1);
        const int r  = idx / 33;
        const int c4 = (idx - r * 33) * 4;
        int gy = gy0 + r;  gy = (gy < HH - 1) ? gy : (HH - 1);
        int gx = gx0 + c4; gx = (gx < WW - 4) ? gx : (WW - 4);
        stage_b128(board + (size_t)gy * WW + gx, &tile[r][c4]);
    }
#if defined(USE_ASYNC_LDS)
    __builtin_amdgcn_s_wait_asynccnt(0);
#endif
    __syncthreads();

    // ---- Output pointers (concatenated flat, reference return order).
    const size_t N1 = (size_t)OW * OW;
    const size_t N2 = (size_t)EW * EW;
    float* __restrict__ bamboo_b = out;
    float* __restrict__ bamboo_w = out + N1;
    float* __restrict__ tiger_b  = out + 2 * N1;
    float* __restrict__ tiger_w  = out + 3 * N1;
    float* __restrict__ tri_b    = out + 4 * N1;
    float* __restrict__ tri_w    = out + 5 * N1;
    float* __restrict__ eye_b    = out + 6 * N1;
    float* __restrict__ eye_w    = out + 6 * N1 + N2;

    // Thread micro-tile: 4 consecutive x (lane-contiguous: wave stores 512B/instr),
    // 2 consecutive y rows. 32 lanes x 4 = 128 x; 8 thread-rows x 2 = 16 y.
    const int xq  = (tid & 31) * 4;   // 0..124, 16B-aligned in LDS
    const int ly0 = (tid >> 5) * 2;   // 0..14
    const int x0  = gx0 + xq;

    // Register window: 4 rows x 6 cols of the LDS tile (vectorizes to ds_load_b128+b64).
    int w[4][6];
#pragma unroll
    for (int rr = 0; rr < 4; ++rr)
#pragma unroll
        for (int cc = 0; cc < 6; ++cc)
            w[rr][cc] = tile[ly0 + rr][xq + cc];

#pragma unroll
    for (int t = 0; t < 2; ++t) {
        const int y = gy0 + ly0 + t;

        f4 vBamB, vBamW, vTigB, vTigW, vTrB, vTrW, vEyeB, vEyeW;
#pragma unroll
        for (int j = 0; j < 4; ++j) {
            // 2x2 corners: A=(y,x) B=(y,x+1) C=(y+1,x) D=(y+1,x+1)
            const int A = w[t][j],     B = w[t][j + 1];
            const int C = w[t + 1][j], D = w[t + 1][j + 1];
            const int bs = (A == 1) + (B == 1) + (C == 1) + (D == 1);
            const int ws = (A == 2) + (B == 2) + (C == 2) + (D == 2);
            const int es = (A == 0) + (B == 0) + (C == 0) + (D == 0);

            const bool bamB = ((A == 1) & (D == 1) & (B == 0) & (C == 0)) |
                              ((B == 1) & (C == 1) & (A == 0) & (D == 0));
            const bool bamW = ((A == 2) & (D == 2) & (B == 0) & (C == 0)) |
                              ((B == 2) & (C == 2) & (A == 0) & (D == 0));
            const bool anyE = es >= 1;
            vBamB[j] = bamB ? 1.0f : 0.0f;
            vBamW[j] = bamW ? 1.0f : 0.0f;
            vTigB[j] = ((bs >= 2) & anyE) ? 1.0f : 0.0f;
            vTigW[j] = ((ws >= 2) & anyE) ? 1.0f : 0.0f;
            vTrB[j]  = ((bs == 3) & (es == 1)) ? 1.0f : 0.0f;
            vTrW[j]  = ((ws == 3) & (es == 1)) ? 1.0f : 0.0f;

            // Eye at (y,x): center (y+1,x+1)==empty, cross all one color.
            const int ctr = w[t + 1][j + 1];
            const int nn  = w[t][j + 1],     ss = w[t + 2][j + 1];
            const int wl  = w[t + 1][j],     ee = w[t + 1][j + 2];
            vEyeB[j] = ((ctr == 0) & (nn == 1) & (ss == 1) & (wl == 1) & (ee == 1)) ? 1.0f : 0.0f;
            vEyeW[j] = ((ctr == 0) & (nn == 2) & (ss == 2) & (wl == 2) & (ee == 2)) ? 1.0f : 0.0f;
        }

        if (y < OW) {
            const size_t o = (size_t)y * OW + x0;
            if (x0 + 3 < OW) {  // full quad in-range: one b128 NT store per map
                __builtin_nontemporal_store(vBamB, (uf4*)(bamboo_b + o));
                __builtin_nontemporal_store(vBamW, (uf4*)(bamboo_w + o));
                __builtin_nontemporal_store(vTigB, (uf4*)(tiger_b + o));
                __builtin_nontemporal_store(vTigW, (uf4*)(tiger_w + o));
                __builtin_nontemporal_store(vTrB,  (uf4*)(tri_b + o));
                __builtin_nontemporal_store(vTrW,  (uf4*)(tri_w + o));
            } else {            // ragged right edge: guarded scalar stores
#pragma unroll
                for (int j = 0; j < 4; ++j)
                    if (x0 + j < OW) {
                        __builtin_nontemporal_store(vBamB[j], bamboo_b + o + j);
                        __builtin_nontemporal_store(vBamW[j], bamboo_w + o + j);
                        __builtin_nontemporal_store(vTigB[j], tiger_b + o + j);
                        __builtin_nontemporal_store(vTigW[j], tiger_w + o + j);
                        __builtin_nontemporal_store(vTrB[j],  tri_b + o + j);
                        __builtin_nontemporal_store(vTrW[j],  tri_w + o + j);
                    }
            }
        }
        if (y < EW) {
            const size_t o = (size_t)y * EW + x0;
            if (x0 + 3 < EW) {
                __builtin_nontemporal_store(vEyeB, (uf4*)(eye_b + o));
                __builtin_nontemporal_store(vEyeW, (uf4*)(eye_w + o));
            } else {
#pragma unroll
                for (int j = 0; j < 4; ++j)
                    if (x0 + j < EW) {
                        __builtin_nontemporal_store(vEyeB[j], eye_b + o + j);
                        __builtin_nontemporal_store(vEyeW[j], eye_w + o + j);
                    }
            }
        }
    }
}

extern "C" void kernel_launch(void* const* d_in, const int* in_sizes, int n_in,
                              void* d_out, int out_size, void* d_ws, size_t ws_size,
                              hipStream_t stream) {
    const int* board = (const int*)d_in[0];
    float* out = (float*)d_out;
    dim3 grid((OW + TX - 1) / TX, (OW + TY - 1) / TY); // 32 x 256 blocks
    motif_stencil_kernel<<<grid, 256, 0, stream>>>(board, out);
}